// predictor_interp2d_11175504904480
// MI455X (gfx1250) — compile-verified
//
#include <hip/hip_runtime.h>

typedef __attribute__((ext_vector_type(2))) float v2f;
typedef __attribute__((ext_vector_type(8))) float v8f;

#define GH   256
#define GW   256
#define NPTS 1024
#define NCH  4

// Block: 256 threads = 8 waves. Each wave computes argmin for a strip of 16
// grid cells over all 1024 points using V_WMMA_F32_16X16X4_F32 tiles.
//
// B-operand LDS layout: one 8-byte record per (half, point):
//   s_B[n]        = (-2*px, -2*py)   -> lanes 0-15  (K0, K1)
//   s_B[NPTS + n] = (px^2+py^2, 0)   -> lanes 16-31 (K2, K3)
// so every lane issues exactly one ds_load_b64 per tile, no divergence.
__global__ __launch_bounds__(256) void nn_gather_wmma_kernel(
    const float* __restrict__ R_pc,   // [B, C, N]
    const float* __restrict__ XY_pc,  // [B, 2, N]
    float* __restrict__ out)          // [B, C, H*W]
{
    __shared__ v2f s_B[2 * NPTS];
    __shared__ int s_idx[8 * 16];

    const int b   = blockIdx.y;
    const int tid = threadIdx.x;

    // ---- stage point terms into LDS (whole block, coalesced) ----
    const float* XYb = XY_pc + (size_t)b * 2 * NPTS;
    for (int i = tid; i < NPTS; i += 256) {
        float px = XYb[i];
        float py = XYb[NPTS + i];
        v2f lo, hi2;
        lo.x  = -2.0f * px;
        lo.y  = -2.0f * py;
        hi2.x = px * px + py * py;
        hi2.y = 0.0f;
        s_B[i]        = lo;
        s_B[NPTS + i] = hi2;
    }
    __syncthreads();

    const int wave = tid >> 5;        // 0..7
    const int lane = tid & 31;
    const int lm   = lane & 15;
    const int hi   = lane >> 4;       // 0: lanes 0-15 (K0,K1), 1: lanes 16-31 (K2,K3)

    const int strip = blockIdx.x * 128 + wave * 16;   // first grid cell of strip

    // A matrix 16x4 (MxK): row M = grid cell; K-vector = [gx, gy, 1, 0]
    int   g  = strip + lm;
    float gx = ((float)(g & (GW - 1)) + 0.5f) * (1.0f / GW);
    float gy = ((float)(g >> 8)       + 0.5f) * (1.0f / GH);
    v2f a;
    a.x = hi ? 1.0f : gx;
    a.y = hi ? 0.0f : gy;

    float best[8];
    int   bidx[8];
#pragma unroll
    for (int r = 0; r < 8; ++r) { best[r] = 3.4e38f; bidx[r] = 0; }

    // ---- main loop: 64 WMMA tiles across N ----
    const v2f* bp = &s_B[hi * NPTS + lm];
#pragma unroll 4
    for (int t = 0; t < NPTS / 16; ++t) {
        v2f bv = bp[t * 16];          // single ds_load_b64, uniform control flow

        v8f c = {};
        c = __builtin_amdgcn_wmma_f32_16x16x4_f32(
                /*neg_a=*/false, a, /*neg_b=*/false, bv,
                /*c_mod=*/(short)0, c, /*reuse_a=*/false, /*reuse_b=*/false);

        int n = t * 16 + lm;
        // lane holds d2[M = r + 8*hi, N = n] in c[r]
#pragma unroll
        for (int r = 0; r < 8; ++r) {
            float v  = c[r];
            bool  lt = v < best[r];
            best[r]  = lt ? v : best[r];
            bidx[r]  = lt ? n : bidx[r];
        }
    }

    // ---- argmin butterfly over the 16 lanes of each half (same M rows) ----
#pragma unroll
    for (int m = 8; m >= 1; m >>= 1) {
#pragma unroll
        for (int r = 0; r < 8; ++r) {
            float ov = __shfl_xor(best[r], m, 16);
            int   oi = __shfl_xor(bidx[r], m, 16);
            // branchless first-min tie-break (matches jnp.argmin semantics)
            int take = (int)(ov < best[r]) |
                       ((int)(ov == best[r]) & (int)(oi < bidx[r]));
            best[r] = take ? ov : best[r];
            bidx[r] = take ? oi : bidx[r];
        }
    }

    // one lane per half publishes the 8 row indices (rows M = hi*8 + r)
    if (lm == 0) {
#pragma unroll
        for (int r = 0; r < 8; ++r)
            s_idx[wave * 16 + hi * 8 + r] = bidx[r];
    }
    __syncthreads();

    // ---- gather channels + store: 16 rows x 4 ch = 64 outputs per wave ----
    const float* Rb = R_pc + (size_t)b * NCH * NPTS;
    float*       ob = out  + (size_t)b * NCH * (GH * GW);
#pragma unroll
    for (int j = 0; j < 2; ++j) {
        int e   = j * 32 + lane;   // 0..63
        int row = e >> 2;          // 0..15
        int ch  = e & 3;
        int idx = s_idx[wave * 16 + row];
        ob[(size_t)ch * (GH * GW) + strip + row] = Rb[ch * NPTS + idx];
    }
}

extern "C" void kernel_launch(void* const* d_in, const int* in_sizes, int n_in,
                              void* d_out, int out_size, void* d_ws, size_t ws_size,
                              hipStream_t stream) {
    const float* R_pc  = (const float*)d_in[0];   // [B, C, N] f32
    const float* XY_pc = (const float*)d_in[1];   // [B, 2, N] f32
    float* out = (float*)d_out;                   // [B, C, H, W] f32

    int B = in_sizes[1] / (2 * NPTS);             // = 2 per setup_inputs
    dim3 grid((GH * GW) / 128, B);
    dim3 block(256);
    nn_gather_wmma_kernel<<<grid, block, 0, stream>>>(R_pc, XY_pc, out);
}